// SoftmaxMatcher_37185826849270
// MI455X (gfx1250) — compile-verified
//
#include <hip/hip_runtime.h>

typedef float v2f __attribute__((ext_vector_type(2)));
typedef float v8f __attribute__((ext_vector_type(8)));

#define BQ   4
#define CCH  256
#define NKP  512
#define HW   65536
#define WDIM 256

// ---------------------------------------------------------------------------
// Pass 1a: L2-normalize src keypoint descriptors, store transposed [q][n][c]
// so WMMA A-fragments are contiguous float2 loads per lane.
// ---------------------------------------------------------------------------
__global__ void k_src_norm(const float* __restrict__ kd, float* __restrict__ src_hat) {
    int q = blockIdx.x >> 9;          // 512 blocks per q
    int n = blockIdx.x & 511;
    int c = threadIdx.x;
    float x = kd[((size_t)(2 * q) * CCH + c) * NKP + n];
    __shared__ float red[256];
    red[c] = x * x;
    __syncthreads();
    for (int s = 128; s > 0; s >>= 1) {
        if (c < s) red[c] += red[c + s];
        __syncthreads();
    }
    float inv = 1.0f / fmaxf(sqrtf(red[0]), 1e-12f);
    src_hat[((size_t)q * NKP + n) * CCH + c] = x * inv;
}

// ---------------------------------------------------------------------------
// Pass 1b: per-pixel inverse L2 norm of target dense descriptors (normalize
// on the fly later: dot(src_hat, tgt_hat) = dot(src_hat, tgt) * invn[m]).
// ---------------------------------------------------------------------------
__global__ void k_tgt_invnorm(const float* __restrict__ dd, float* __restrict__ invn) {
    int q = blockIdx.x >> 8;          // 256 blocks per q
    int m = ((blockIdx.x & 255) << 8) + threadIdx.x;
    const float* p = dd + (size_t)((2 * q + 1) * CCH) * HW + m;
    float s = 0.f;
    #pragma unroll 8
    for (int c = 0; c < CCH; ++c) {
        float x = p[(size_t)c * HW];
        s = fmaf(x, x, s);
    }
    invn[(size_t)q * HW + m] = 1.0f / fmaxf(sqrtf(s), 1e-12f);
}

// ---------------------------------------------------------------------------
// Pass 2: fused GEMM (f32 WMMA 16x16x4) + fixed-shift softmax + expected
// coordinates. One WG = 16 keypoint rows of one batch; 8 waves stride over
// the 4096 column tiles; A tile lives in registers for the whole kernel.
// B fragments are software-pipelined through two 16-fragment register
// buffers so load latency overlaps the serialized WMMA accumulate chain.
// ---------------------------------------------------------------------------

// Issue the 32 b32 loads for fragment group g (K channels 64*g .. 64*g+63).
#define LOADG(buf, g)                                                        \
    _Pragma("unroll")                                                        \
    for (int kk = 0; kk < 16; ++kk) {                                        \
        const float* bk = tgt + boff + (size_t)(4 * ((g) * 16 + kk)) * HW;   \
        buf[kk].x = bk[0];                                                   \
        buf[kk].y = bk[HW];                                                  \
    }

// Run the 16 chained WMMAs for fragment group g.
#define WMMAG(buf, g)                                                        \
    _Pragma("unroll")                                                        \
    for (int kk = 0; kk < 16; ++kk) {                                        \
        acc = __builtin_amdgcn_wmma_f32_16x16x4_f32(                         \
            false, afrag[(g) * 16 + kk], false, buf[kk], (short)0, acc,      \
            false, false);                                                   \
    }

__launch_bounds__(256)
__global__ void k_match(const float* __restrict__ src_hat,
                        const float* __restrict__ dd,
                        const float* __restrict__ invn,
                        float* __restrict__ out_coords) {
    int q   = blockIdx.x >> 5;
    int n0  = (blockIdx.x & 31) << 4;
    int tid = threadIdx.x;
    int wid = tid >> 5;
    int lane = tid & 31;
    int hi  = lane >> 4;              // half-wave selects K pair / row half
    int l15 = lane & 15;

    // A fragments: 16 rows x 256 channels, 64 x v2f per lane (kept in VGPRs).
    // Layout per ISA: lane l (l<16): K = {4k, 4k+1}; lane l+16: K = {4k+2, 4k+3}.
    v2f afrag[64];
    const v2f* arow = (const v2f*)(src_hat + ((size_t)q * NKP + n0 + l15) * CCH);
    #pragma unroll
    for (int k = 0; k < 64; ++k) afrag[k] = arow[2 * k + hi];

    const float* tgt = dd + (size_t)((2 * q + 1) * CCH) * HW;  // uniform base
    const float* inq = invn + (size_t)q * HW;

    float se[8], su[8], sv[8];
    #pragma unroll
    for (int j = 0; j < 8; ++j) { se[j] = 0.f; su[j] = 0.f; sv[j] = 0.f; }

    for (int t = wid; t < HW / 16; t += 8) {
        int col  = t * 16 + l15;          // this lane's pixel column
        int boff = 2 * hi * HW + col;     // per-lane 32-bit element offset

        v2f bA[16], bB[16];
        v8f acc = {};
        // 2-deep ping-pong pipeline over 4 groups of 16 K-fragments.
        LOADG(bA, 0)
        LOADG(bB, 1)
        WMMAG(bA, 0)
        LOADG(bA, 2)
        WMMAG(bB, 1)
        LOADG(bB, 3)
        WMMAG(bA, 2)
        WMMAG(bB, 3)

        // acc[j] = dot for row (j + 8*hi) at column `col`
        float wn = 100.0f * inq[col];             // 1/temp * inv-norm
        float uu = (float)(col & (WDIM - 1));
        float vv = (float)(col >> 8);
        #pragma unroll
        for (int j = 0; j < 8; ++j) {
            // fixed shift: logits in [-100,100]; dominant term stays normal f32
            float p = __expf(fmaf(acc[j], wn, -100.0f));
            se[j] += p;
            su[j] = fmaf(p, uu, su[j]);
            sv[j] = fmaf(p, vv, sv[j]);
        }
    }

    // Reduce over the 16 lanes of each half-wave (xor masks stay in-half).
    #pragma unroll
    for (int j = 0; j < 8; ++j) {
        #pragma unroll
        for (int off = 1; off < 16; off <<= 1) {
            se[j] += __shfl_xor(se[j], off, 32);
            su[j] += __shfl_xor(su[j], off, 32);
            sv[j] += __shfl_xor(sv[j], off, 32);
        }
    }

    // Deterministic cross-wave combine (fixed summation order, no FP atomics).
    __shared__ float lse[8][16], lsu[8][16], lsv[8][16];
    if (l15 == 0) {
        #pragma unroll
        for (int j = 0; j < 8; ++j) {
            lse[wid][j + 8 * hi] = se[j];
            lsu[wid][j + 8 * hi] = su[j];
            lsv[wid][j + 8 * hi] = sv[j];
        }
    }
    __syncthreads();
    if (tid < 16) {
        float S = 0.f, U = 0.f, V = 0.f;
        #pragma unroll
        for (int w = 0; w < 8; ++w) { S += lse[w][tid]; U += lsu[w][tid]; V += lsv[w][tid]; }
        S = fmaxf(S, 1e-30f);
        size_t o = ((size_t)q * NKP + n0 + tid) * 2;
        out_coords[o]     = U / S;
        out_coords[o + 1] = V / S;
    }
}

// ---------------------------------------------------------------------------
// Pass 3: bilinear sample (align_corners, clipped) of target scores and
// normalized target descriptors at pseudo-coords; match weights.
// ---------------------------------------------------------------------------
__global__ void k_weights(const float* __restrict__ kscores,
                          const float* __restrict__ sdense,
                          const float* __restrict__ dd,
                          const float* __restrict__ invn,
                          const float* __restrict__ src_hat,
                          float* __restrict__ dout) {
    int p = blockIdx.x;
    int q = p >> 9;
    int n = p & 511;
    int c = threadIdx.x;

    float u = dout[(size_t)p * 2];
    float v = dout[(size_t)p * 2 + 1];
    u = fminf(fmaxf(u, 0.f), 255.f);
    v = fminf(fmaxf(v, 0.f), 255.f);
    float u0 = floorf(u), v0 = floorf(v);
    float u1 = fminf(u0 + 1.f, 255.f), v1 = fminf(v0 + 1.f, 255.f);
    float wu = u - u0, wv = v - v0;
    int m00 = (int)v0 * WDIM + (int)u0;
    int m01 = (int)v0 * WDIM + (int)u1;
    int m10 = (int)v1 * WDIM + (int)u0;
    int m11 = (int)v1 * WDIM + (int)u1;
    float w00 = (1.f - wv) * (1.f - wu), w01 = (1.f - wv) * wu;
    float w10 = wv * (1.f - wu),         w11 = wv * wu;

    const float* tb = dd + ((size_t)((2 * q + 1) * CCH) + c) * HW;
    const float* iq = invn + (size_t)q * HW;
    float tv = w00 * tb[m00] * iq[m00] + w01 * tb[m01] * iq[m01]
             + w10 * tb[m10] * iq[m10] + w11 * tb[m11] * iq[m11];
    float contrib = src_hat[((size_t)q * NKP + n) * CCH + c] * tv;

    __shared__ float red[256];
    red[c] = contrib;
    __syncthreads();
    for (int s = 128; s > 0; s >>= 1) {
        if (c < s) red[c] += red[c + s];
        __syncthreads();
    }
    if (c == 0) {
        const float* sb = sdense + (size_t)(2 * q + 1) * HW;
        float ps  = w00 * sb[m00] + w01 * sb[m01] + w10 * sb[m10] + w11 * sb[m11];
        float dms = red[0] * (1.0f / CCH);
        float ss  = kscores[(size_t)(2 * q) * NKP + n];
        dout[(size_t)BQ * NKP * 2 + (size_t)q * NKP + n] = 0.5f * (dms + 1.0f) * ss * ps;
    }
}

// ---------------------------------------------------------------------------
extern "C" void kernel_launch(void* const* d_in, const int* in_sizes, int n_in,
                              void* d_out, int out_size, void* d_ws, size_t ws_size,
                              hipStream_t stream) {
    const float* kscores = (const float*)d_in[0];  // [8,1,512]
    const float* kdesc   = (const float*)d_in[1];  // [8,256,512]
    const float* sdense  = (const float*)d_in[2];  // [8,1,256,256]
    const float* ddense  = (const float*)d_in[3];  // [8,256,256,256]
    float* out = (float*)d_out;                    // 4096 coords + 2048 weights

    float* src_hat = (float*)d_ws;                 // 4*512*256 floats (2 MB)
    float* invn    = src_hat + (size_t)BQ * NKP * CCH;  // 4*65536 floats (1 MB)

    k_src_norm  <<<BQ * NKP,        256, 0, stream>>>(kdesc, src_hat);
    k_tgt_invnorm<<<BQ * HW / 256,  256, 0, stream>>>(ddense, invn);
    k_match     <<<BQ * (NKP / 16), 256, 0, stream>>>(src_hat, ddense, invn, out);
    k_weights   <<<BQ * NKP,        256, 0, stream>>>(kscores, sdense, ddense, invn, src_hat, out);
}